// ConvCaps2D_3891240370332
// MI455X (gfx1250) — compile-verified
//
#include <hip/hip_runtime.h>

typedef float v2f __attribute__((ext_vector_type(2)));
typedef float v8f __attribute__((ext_vector_type(8)));

#define HO 62
#define WOUT 62
#define PTOT (HO * WOUT)   // 3844 spatial positions per batch
#define HI 64
#define WI 64
#define TI 8
#define DI 8
#define TO 16
#define DOUT 8
#define CG (TO * DOUT)     // 128 channels per group
#define KDIM 72            // DI*3*3
#define NBATCH 8

// ---------------------------------------------------------------------------
// Kernel 1: grouped conv as WMMA GEMM with async-to-LDS im2col staging.
//   Per group g: U[p, c] = sum_k Im2col[p, k] * W[c, k]
//   Tile: M=16 spatial positions x N=16 channels, K=72 in 18 steps of 4
//   via V_WMMA_F32_16X16X4_F32 (f32 in / f32 acc, matches reference dtype).
//
// The 16-position im2col tile (72 x 16 f32, k-major) is shared by all 8 waves
// of the block -> staged once into LDS with GLOBAL_LOAD_ASYNC_TO_LDS_B32
// (ASYNCcnt), then each wave reads A with immediate-offset ds_load_b32:
//   A[m=l16][k], lanes<16 carry K={k0,k0+1}, lanes>=16 K={k0+2,k0+3}
//   LDS addr = (k*16 + l16)*4 = lane_base + t*256  (bank-conflict free).
// B (4x16): col n=l16, same K split; preloaded to 18 v2f regs per lane.
// C/D: VGPR r -> M=r (lanes<16) / r+8 (lanes>=16), N = lane&15.
// grid = (ceil(PTOT/16), TI, N); block = 256 (8 waves; wave w -> ctile w)
// ---------------------------------------------------------------------------
__global__ __launch_bounds__(256) void conv_caps_wmma_kernel(
    const float* __restrict__ x, const float* __restrict__ w,
    const float* __restrict__ cbias, float* __restrict__ uhat)
{
  __shared__ float As[1280];    // 72*16 = 1152 used, padded to 5*256

  const int lane = threadIdx.x & 31;
  const int wave = threadIdx.x >> 5;
  const int half = lane >> 4;   // selects K pair {0,1} vs {2,3}
  const int l16  = lane & 15;
  const int g  = blockIdx.y;    // input capsule type (conv group)
  const int n  = blockIdx.z;    // batch
  const int p0 = blockIdx.x * 16;
  const int ctile = wave;       // 16-channel slice of the group's 128

  // ---- cooperative async fill of the im2col A-tile: As[k][m] ----
  const float* xg = x + ((size_t)n * TI + g) * (DI * HI * WI);
  {
    const int tid = threadIdx.x;
#pragma unroll
    for (int it = 0; it < 5; ++it) {
      const int e = tid + it * 256;          // 0..1279 (tail clamped, no EXEC holes)
      int k = e >> 4; if (k > KDIM - 1) k = KDIM - 1;
      const int m = e & 15;
      int pp = p0 + m; if (pp >= PTOT) pp = PTOT - 1;
      const int ho = pp / WOUT, wo = pp - ho * WOUT;
      const int di = k / 9, r9 = k - di * 9, kh = r9 / 3, kw = r9 - kh * 3;
      const float* ga = xg + (di * HI + ho + kh) * WI + (wo + kw);
      const unsigned ldsOff = (unsigned)(uintptr_t)(&As[e]);  // LDS byte offset
      asm volatile("global_load_async_to_lds_b32 %0, %1, off"
                   :: "v"(ldsOff), "v"(ga) : "memory");
    }
    asm volatile("s_wait_asynccnt 0" ::: "memory");
  }

  // ---- preload B (weights) for (g, ctile): one v2f per K-step ----
  v2f breg[18];
  {
    const float* wp =
        w + ((size_t)(g * CG + ctile * 16 + l16)) * KDIM + half * 2;
#pragma unroll
    for (int t = 0; t < 18; ++t) breg[t] = *(const v2f*)(wp + t * 4);
  }

  __syncthreads();   // A-tile visible to all waves

  // ---- 18-step WMMA accumulation, A from LDS at immediate offsets ----
  const float* Ap = &As[(half << 5) + l16];   // (half*2)*16 + l16
  v8f acc = {};
#pragma unroll
  for (int t = 0; t < 18; ++t) {
    v2f a;
    a.x = Ap[t * 64];        // k = t*4 + half*2
    a.y = Ap[t * 64 + 16];   // k+1
    acc = __builtin_amdgcn_wmma_f32_16x16x4_f32(
        /*neg_a=*/false, a, /*neg_b=*/false, breg[t],
        /*c_mod=*/(short)0, acc, /*reuse_a=*/false, /*reuse_b=*/false);
  }

  // ---- store D (+ conv bias); u_hat layout (n, p, ti, to, do) ----
  const int ch   = g * CG + ctile * 16 + l16;   // channel across all groups
  const float cb = cbias[ch];
#pragma unroll
  for (int r = 0; r < 8; ++r) {
    const int m  = r + half * 8;   // spatial row within tile
    const int pp = p0 + m;
    if (pp < PTOT)
      uhat[((size_t)n * PTOT + pp) * (TI * CG) + ch] = acc[r] + cb;
  }
}

// ---------------------------------------------------------------------------
// Kernel 2: dynamic routing, one wave32 per (n, p) position; all 3 rounds in
// registers. Lane map: to = lane&15, ti group = lane>>4 (4 tis per lane).
// Softmax over the 16 output types = shfl_xor reduction within a half-wave;
// Ti-sum = local 4-term sum + xor-16 partner exchange.
// ---------------------------------------------------------------------------
__global__ __launch_bounds__(256) void routing_kernel(
    const float* __restrict__ uhat, const float* __restrict__ rbias,
    float* __restrict__ out)
{
  const int lane = threadIdx.x & 31;
  const int wave = threadIdx.x >> 5;
  const int gid  = blockIdx.x * 8 + wave;   // flat (n, p)
  const int n = gid / PTOT;
  const int p = gid - n * PTOT;
  const int to  = lane & 15;
  const int tig = lane >> 4;                // ti = tig*4 + j

  const float* ub = uhat + ((size_t)n * PTOT + p) * (TI * CG);
  float uh[4][8];
#pragma unroll
  for (int j = 0; j < 4; ++j) {
    const float4* q = (const float4*)(ub + (tig * 4 + j) * CG + to * DOUT);
    const float4 lo = q[0], hi = q[1];
    uh[j][0] = lo.x; uh[j][1] = lo.y; uh[j][2] = lo.z; uh[j][3] = lo.w;
    uh[j][4] = hi.x; uh[j][5] = hi.y; uh[j][6] = hi.z; uh[j][7] = hi.w;
  }
  float rb[8];
#pragma unroll
  for (int d = 0; d < 8; ++d) rb[d] = rbias[to * DOUT + d];

  float b[4] = {0.f, 0.f, 0.f, 0.f};
  float v[8];

  for (int r = 0; r < 3; ++r) {
    // coupling coefficients: softmax of b over the 16 'to' lanes, per ti j
    float c[4];
#pragma unroll
    for (int j = 0; j < 4; ++j) {
      float mx = b[j];
#pragma unroll
      for (int mask = 1; mask <= 8; mask <<= 1)
        mx = fmaxf(mx, __shfl_xor(mx, mask, 32));
      const float e = __expf(b[j] - mx);
      float s = e;
#pragma unroll
      for (int mask = 1; mask <= 8; mask <<= 1)
        s += __shfl_xor(s, mask, 32);
      c[j] = e / s;
    }
    // s[d] = sum_ti c*u_hat + routing_bias; then squash over do
    float n2 = 0.f;
#pragma unroll
    for (int d = 0; d < 8; ++d) {
      float t = c[0] * uh[0][d] + c[1] * uh[1][d] +
                c[2] * uh[2][d] + c[3] * uh[3][d];
      t += __shfl_xor(t, 16, 32);   // combine the two ti half-waves
      t += rb[d];
      v[d] = t;
      n2 += t * t;
    }
    const float scale = n2 / ((1.f + n2) * sqrtf(n2 + 1e-9f));
#pragma unroll
    for (int d = 0; d < 8; ++d) v[d] *= scale;

    if (r < 2) {  // agreement update: b += <u_hat, v>
#pragma unroll
      for (int j = 0; j < 4; ++j) {
        float a = 0.f;
#pragma unroll
        for (int d = 0; d < 8; ++d) a += uh[j][d] * v[d];
        b[j] += a;
      }
    }
  }

  // Output layout (N, To, Do, Ho, Wo); both halves hold identical v — store once.
  if (tig == 0) {
#pragma unroll
    for (int d = 0; d < 8; ++d)
      out[((size_t)(n * TO + to) * DOUT + d) * PTOT + p] = v[d];
  }
}

extern "C" void kernel_launch(void* const* d_in, const int* in_sizes, int n_in,
                              void* d_out, int out_size, void* d_ws, size_t ws_size,
                              hipStream_t stream) {
  const float* x      = (const float*)d_in[0];
  const float* conv_w = (const float*)d_in[1];
  const float* conv_b = (const float*)d_in[2];
  const float* rbias  = (const float*)d_in[3];
  float* uhat = (float*)d_ws;          // N*PTOT*1024 f32 ~= 126 MB scratch
  float* out  = (float*)d_out;

  dim3 gconv((PTOT + 15) / 16, TI, NBATCH);
  conv_caps_wmma_kernel<<<gconv, 256, 0, stream>>>(x, conv_w, conv_b, uhat);

  const int nwaves = NBATCH * PTOT;    // 30752, divisible by 8
  routing_kernel<<<nwaves / 8, 256, 0, stream>>>(uhat, rbias, out);
}